// GATNet_2336462209634
// MI455X (gfx1250) — compile-verified
//
#include <hip/hip_runtime.h>

typedef __attribute__((ext_vector_type(2))) float v2f;
typedef __attribute__((ext_vector_type(8))) float v8f;

#define NEG_SLOPE 0.2f

__device__ __forceinline__ void atomicMaxF(float* addr, float val) {
  // Monotonic-bit-pattern trick: int-max for non-negative, uint-min for negative.
  if (val >= 0.0f) {
    atomicMax((int*)addr, __float_as_int(val));
  } else {
    atomicMin((unsigned int*)addr, __float_as_uint(val));
  }
}

// ---------------------------------------------------------------------------
// C[M,Ncols] = A[M,128] @ B[128,Ncols], f32, via V_WMMA_F32_16X16X4_F32.
// One 16x16 output tile per wave32; K=128 -> 32 WMMA issues per tile.
// A layout (32-bit 16x4): lanes 0-15 = rows, VGPR0/1 = K0/K1; lanes 16-31 = K2/K3.
// B layout (32-bit 4x16): lanes = cols; VGPR0/1 = K0/K1 (low half), K2/K3 (high half).
// C layout: VGPR v -> row v (lanes 0-15) / v+8 (lanes 16-31), col = lane&15.
// ---------------------------------------------------------------------------
__global__ void __launch_bounds__(128) gat_gemm_wmma_f32(
    const float* __restrict__ A, const float* __restrict__ B,
    float* __restrict__ C, int M, int Ncols, int ldb, int ldc) {
  const int lane   = threadIdx.x & 31;
  const int wave   = threadIdx.x >> 5;
  const int tile_m = blockIdx.x;
  const int tile_n = blockIdx.y * 4 + wave;
  if (tile_n * 16 >= Ncols) return;          // wave-uniform exit: EXEC stays all-1s
  const int half = lane >> 4;                // 0: lanes 0-15, 1: lanes 16-31
  const int kb   = half * 2;                 // K sub-offset owned by this half
  const int mrow = tile_m * 16 + (lane & 15);
  const int col  = tile_n * 16 + (lane & 15);
  // Wave-uniform interior test: all operands uniform, so no divergence.
  const bool full = (tile_m * 16 + 16 <= M) && (tile_n * 16 + 16 <= Ncols);
  v8f acc = {};
  if (full) {
    // Fast path: no guards, A as 8B vector loads.
    const v2f*   Ap = (const v2f*)(A + (size_t)mrow * 128 + kb);
    const float* Bp = B + (size_t)kb * ldb + col;
#pragma unroll 4
    for (int k = 0; k < 32; ++k) {           // 32 K-steps of 4
      v2f a = Ap[k * 2];                     // floats [k*4+kb, k*4+kb+1]
      v2f b;
      b.x = Bp[(size_t)(k * 4) * ldb];
      b.y = Bp[(size_t)(k * 4 + 1) * ldb];
      acc = __builtin_amdgcn_wmma_f32_16x16x4_f32(false, a, false, b,
                                                  (short)0, acc, false, false);
    }
  } else {
    // Edge tile: clamp addresses, zero via multiply (EXEC stays full).
    const int mrow_c = mrow < M ? mrow : M - 1;
    const int col_c  = col < Ncols ? col : Ncols - 1;
    const float a_ok = (mrow < M) ? 1.0f : 0.0f;
    const float b_ok = (col < Ncols) ? 1.0f : 0.0f;
    const float* Ap = A + (size_t)mrow_c * 128;
    const float* Bp = B + col_c;
#pragma unroll 4
    for (int k = 0; k < 128; k += 4) {
      v2f a, b;
      a.x = Ap[k + kb] * a_ok;
      a.y = Ap[k + kb + 1] * a_ok;
      b.x = Bp[(size_t)(k + kb) * ldb] * b_ok;
      b.y = Bp[(size_t)(k + kb + 1) * ldb] * b_ok;
      acc = __builtin_amdgcn_wmma_f32_16x16x4_f32(false, a, false, b,
                                                  (short)0, acc, false, false);
    }
  }
  const int rbase = tile_m * 16 + half * 8;
  float* Cp = C + (size_t)rbase * ldc + col;
  if (full) {
#pragma unroll
    for (int v = 0; v < 8; ++v) Cp[(size_t)v * ldc] = acc[v];
  } else {
#pragma unroll
    for (int v = 0; v < 8; ++v)
      if (rbase + v < M && col < Ncols) Cp[(size_t)v * ldc] = acc[v];
  }
}

// al_src[n,h] = sum_d h[n,h,d]*a_src[h,d]; likewise al_dst. One thread per (n,h).
template <int HEADS, int DIM>
__global__ void gat_compute_al(const float* __restrict__ h,
                               const float* __restrict__ a_src,
                               const float* __restrict__ a_dst,
                               float* __restrict__ als, float* __restrict__ ald,
                               int n) {
  const int t = blockIdx.x * blockDim.x + threadIdx.x;
  if (t >= n * HEADS) return;
  const int hd   = t % HEADS;
  const int node = t / HEADS;
  const float* hp = h + (size_t)node * (HEADS * DIM) + hd * DIM;
  const float* as = a_src + hd * DIM;
  const float* ad = a_dst + hd * DIM;
  float s = 0.0f, d = 0.0f;
#pragma unroll
  for (int j = 0; j < DIM; ++j) {
    const float v = hp[j];
    s += v * as[j];
    d += v * ad[j];
  }
  als[t] = s;
  ald[t] = d;
}

__global__ void gat_fill_f32(float* __restrict__ p, float v, int count) {
  const int t = blockIdx.x * blockDim.x + threadIdx.x;
  if (t < count) p[t] = v;
}

// Segment-max of leaky_relu(al_src[src]+al_dst[dst]) into mx[dst,h].
// One thread per (edge, head); edges >= E are self-loops.
template <int HEADS>
__global__ void gat_edge_max(const long long* __restrict__ ei, int E, int n,
                             const float* __restrict__ als,
                             const float* __restrict__ ald,
                             float* __restrict__ mx) {
  const int t = blockIdx.x * blockDim.x + threadIdx.x;
  if (t >= (E + n) * HEADS) return;
  const int hd = t % HEADS;
  const int e  = t / HEADS;
  int s, d;
  if (e < E) { s = (int)ei[e]; d = (int)ei[E + e]; } else { s = d = e - E; }
  float x = als[(size_t)s * HEADS + hd] + ald[(size_t)d * HEADS + hd];
  x = x > 0.0f ? x : NEG_SLOPE * x;
  atomicMaxF(mx + (size_t)d * HEADS + hd, x);
}

// w = exp(lrelu - mx[dst]); denom[dst,h] += w; acc[dst,h,:] += w * h[src,h,:].
// One thread per (edge, head, chunk). chunk 0 also adds the denominator.
template <int HEADS, int DIM, int NCHUNKS>
__global__ void gat_edge_accumulate(const long long* __restrict__ ei, int E, int n,
                                    const float* __restrict__ als,
                                    const float* __restrict__ ald,
                                    const float* __restrict__ mx,
                                    const float* __restrict__ hfeat,
                                    float* __restrict__ denom,
                                    float* __restrict__ acc) {
  const int t = blockIdx.x * blockDim.x + threadIdx.x;
  if (t >= (E + n) * HEADS * NCHUNKS) return;
  const int ch = t % NCHUNKS;
  const int r  = t / NCHUNKS;
  const int hd = r % HEADS;
  const int e  = r / HEADS;
  int s, d;
  if (e < E) { s = (int)ei[e]; d = (int)ei[E + e]; } else { s = d = e - E; }
  float x = als[(size_t)s * HEADS + hd] + ald[(size_t)d * HEADS + hd];
  x = x > 0.0f ? x : NEG_SLOPE * x;
  const float w = expf(x - mx[(size_t)d * HEADS + hd]);
  if (ch == 0) atomicAdd(denom + (size_t)d * HEADS + hd, w);
  constexpr int ND  = HEADS * DIM;
  constexpr int CSZ = (DIM + NCHUNKS - 1) / NCHUNKS;
  const int j0 = ch * CSZ;
  const int j1 = (j0 + CSZ < DIM) ? (j0 + CSZ) : DIM;
  const float* hp = hfeat + (size_t)s * ND + hd * DIM;
  float*       ap = acc   + (size_t)d * ND + hd * DIM;
  for (int j = j0; j < j1; ++j) atomicAdd(ap + j, w * hp[j]);
}

// out = acc/denom + bias, optional ReLU. In-place on acc.
template <int HEADS, int DIM>
__global__ void gat_finalize(float* __restrict__ acc,
                             const float* __restrict__ denom,
                             const float* __restrict__ bias,
                             int n, int do_relu) {
  constexpr int ND = HEADS * DIM;
  const int t = blockIdx.x * blockDim.x + threadIdx.x;
  if (t >= n * ND) return;
  const int c    = t % ND;
  const int node = t / ND;
  const int hd   = c / DIM;
  float den = denom[(size_t)node * HEADS + hd];
  if (den == 0.0f) den = 1.0f;
  float v = acc[t] / den + bias[c];
  if (do_relu) v = v > 0.0f ? v : 0.0f;
  acc[t] = v;
}

static inline unsigned gat_gdiv(long long a, long long b) {
  return (unsigned)((a + b - 1) / b);
}

extern "C" void kernel_launch(void* const* d_in, const int* in_sizes, int n_in,
                              void* d_out, int out_size, void* d_ws, size_t ws_size,
                              hipStream_t stream) {
  const float*     x      = (const float*)d_in[0];
  const long long* ei     = (const long long*)d_in[1];
  const float*     W1     = (const float*)d_in[2];
  const float*     a_src1 = (const float*)d_in[3];
  const float*     a_dst1 = (const float*)d_in[4];
  const float*     b1     = (const float*)d_in[5];
  const float*     W2     = (const float*)d_in[6];
  const float*     a_src2 = (const float*)d_in[7];
  const float*     a_dst2 = (const float*)d_in[8];
  const float*     b2     = (const float*)d_in[9];

  const int n    = in_sizes[0] / 128;   // 50000 nodes, 128 in-features
  const int E    = in_sizes[1] / 2;     // 800000 edges
  const int F2   = in_sizes[6] / 128;   // 121 out-features (templates assume 121)
  const int Etot = E + n;               // + self loops
  const float NEGINF = -__builtin_inff();

  // Workspace layout (floats)
  float* ws   = (float*)d_ws;
  float* h1   = ws;  ws += (size_t)n * 128;   // layer-1 features x@W1
  float* als1 = ws;  ws += (size_t)n * 8;
  float* ald1 = ws;  ws += (size_t)n * 8;
  float* mx1  = ws;  ws += (size_t)n * 8;
  float* den1 = ws;  ws += (size_t)n * 8;
  float* acc1 = ws;  ws += (size_t)n * 128;   // layer-1 aggregate -> relu(+b1) in place
  float* h2   = ws;  ws += (size_t)n * F2;    // layer-2 features hrelu@W2
  float* als2 = ws;  ws += n;
  float* ald2 = ws;  ws += n;
  float* mx2  = ws;  ws += n;
  float* den2 = ws;  // + n
  float* out  = (float*)d_out;                // doubles as layer-2 accumulator

  const int T = 256;

  // ---------------- Layer 1 (heads=8, dim=16, concat) ----------------
  {
    dim3 grid(gat_gdiv(n, 16), gat_gdiv(gat_gdiv(128, 16), 4));
    gat_gemm_wmma_f32<<<grid, 128, 0, stream>>>(x, W1, h1, n, 128, 128, 128);
  }
  gat_compute_al<8, 16><<<gat_gdiv(n * 8, T), T, 0, stream>>>(h1, a_src1, a_dst1,
                                                              als1, ald1, n);
  gat_fill_f32<<<gat_gdiv(n * 8, T), T, 0, stream>>>(mx1, NEGINF, n * 8);
  gat_fill_f32<<<gat_gdiv(n * 8, T), T, 0, stream>>>(den1, 0.0f, n * 8);
  gat_fill_f32<<<gat_gdiv(n * 128, T), T, 0, stream>>>(acc1, 0.0f, n * 128);
  gat_edge_max<8><<<gat_gdiv((long long)Etot * 8, T), T, 0, stream>>>(
      ei, E, n, als1, ald1, mx1);
  gat_edge_accumulate<8, 16, 1><<<gat_gdiv((long long)Etot * 8, T), T, 0, stream>>>(
      ei, E, n, als1, ald1, mx1, h1, den1, acc1);
  gat_finalize<8, 16><<<gat_gdiv(n * 128, T), T, 0, stream>>>(acc1, den1, b1, n, 1);

  // ---------------- Layer 2 (heads=1, dim=121) ----------------
  {
    dim3 grid(gat_gdiv(n, 16), gat_gdiv(gat_gdiv(F2, 16), 4));
    gat_gemm_wmma_f32<<<grid, 128, 0, stream>>>(acc1, W2, h2, n, F2, F2, F2);
  }
  gat_compute_al<1, 121><<<gat_gdiv(n, T), T, 0, stream>>>(h2, a_src2, a_dst2,
                                                           als2, ald2, n);
  gat_fill_f32<<<gat_gdiv(n, T), T, 0, stream>>>(mx2, NEGINF, n);
  gat_fill_f32<<<gat_gdiv(n, T), T, 0, stream>>>(den2, 0.0f, n);
  gat_fill_f32<<<gat_gdiv(n * F2, T), T, 0, stream>>>(out, 0.0f, n * F2);
  gat_edge_max<1><<<gat_gdiv(Etot, T), T, 0, stream>>>(ei, E, n, als2, ald2, mx2);
  gat_edge_accumulate<1, 121, 8><<<gat_gdiv((long long)Etot * 8, T), T, 0, stream>>>(
      ei, E, n, als2, ald2, mx2, h2, den2, out);
  gat_finalize<1, 121><<<gat_gdiv(n * F2, T), T, 0, stream>>>(out, den2, b2, n, 0);
}